// _RoICrop_53214644798048
// MI455X (gfx1250) — compile-verified
//
#include <hip/hip_runtime.h>

#ifndef __has_builtin
#define __has_builtin(x) 0
#endif

// Pointer element types matching the async-to-LDS builtin signatures.
// b32 variant:  (AS1 int*,  AS3 int*,  imm offset, imm cpol)   [probe-verified]
// b128 variant: (AS1 int4*, AS3 int4*, imm offset, imm cpol)   [per diagnostic]
typedef int v4i __attribute__((vector_size(16)));
typedef __attribute__((address_space(1))) int g_as1_int;
typedef __attribute__((address_space(3))) int l_as3_int;
typedef __attribute__((address_space(1))) v4i g_as1_v4i;
typedef __attribute__((address_space(3))) v4i l_as3_v4i;
typedef __attribute__((address_space(3))) void lds_void_t;

constexpr int R    = 128;
constexpr int C    = 512;
constexpr int H    = 37;
constexpr int W    = 37;
constexpr int GS   = 14;
constexpr int P    = GS * GS;      // 196 grid points per RoI
constexpr int MAP  = H * W;        // 1369 floats per channel map
constexpr int CPB  = 4;            // channels staged per block (contiguous region)
constexpr int MAPS = MAP * CPB;    // 5476 floats = 21904 B, 16B-aligned per block
constexpr int BLK  = 256;          // 8 wave32

// Non-temporal: feat is streamed exactly once (no inter-block reuse).
constexpr int CPOL_NT = 1;         // TH[2:0] = NT

__device__ __forceinline__ int iclamp(int v, int lo, int hi) {
    return v < lo ? lo : (v > hi ? hi : v);
}

__device__ __forceinline__ void async_g2l_b32(const float* g, float* l) {
#if __has_builtin(__builtin_amdgcn_global_load_async_to_lds_b32)
    __builtin_amdgcn_global_load_async_to_lds_b32(
        (g_as1_int*)g, (l_as3_int*)l, /*offset=*/0, /*cpol=*/CPOL_NT);
#else
    unsigned lds_off = (unsigned)(unsigned long long)(lds_void_t*)l;
    asm volatile("global_load_async_to_lds_b32 %0, %1, off"
                 :: "v"(lds_off), "v"(g)
                 : "memory");
#endif
}

// 16B-per-lane async copy: one wave moves 512 contiguous bytes per instruction.
__device__ __forceinline__ void async_g2l_b128(const float* g, float* l) {
#if __has_builtin(__builtin_amdgcn_global_load_async_to_lds_b128)
    __builtin_amdgcn_global_load_async_to_lds_b128(
        (g_as1_v4i*)g, (l_as3_v4i*)l, /*offset=*/0, /*cpol=*/CPOL_NT);
#else
    async_g2l_b32(g + 0, l + 0);
    async_g2l_b32(g + 1, l + 1);
    async_g2l_b32(g + 2, l + 2);
    async_g2l_b32(g + 3, l + 3);
#endif
}

__device__ __forceinline__ void wait_async0() {
#if __has_builtin(__builtin_amdgcn_s_wait_asynccnt)
    __builtin_amdgcn_s_wait_asynccnt(0);
#else
    asm volatile("s_wait_asynccnt 0" ::: "memory");
#endif
}

__global__ __launch_bounds__(BLK)
void roicrop_bilinear_kernel(const float* __restrict__ feat,
                             const float* __restrict__ grid,
                             float* __restrict__ out)
{
    __shared__ __align__(16) float smap[MAPS];  // 4 contiguous channel maps
    __shared__ int4  sidx[P];                   // gathered indices per grid point
    __shared__ float swy[P];
    __shared__ float swx[P];

    const int tid = threadIdx.x;
    const int blocksPerRoi = C / CPB;                 // 128
    const int r  = blockIdx.x / blocksPerRoi;
    const int c0 = (blockIdx.x - r * blocksPerRoi) * CPB;

    // Stage CPB channel maps into LDS with async b128 global->LDS DMA copies.
    // Region base = 21904*m bytes (16B aligned); MAPS = 4*1369 exactly.
    const float* gsrc = feat + (size_t)(r * C + c0) * MAP;
    for (int k = tid; k < MAP; k += BLK) {            // 1369 b128 copies/block
        async_g2l_b128(gsrc + 4 * k, &smap[4 * k]);
    }

    // While the async engine streams the tile, compute the 196 shared sample
    // descriptors (identical across all channels of this RoI).
    if (tid < P) {
        const float2 yx = ((const float2*)grid)[(size_t)r * P + tid]; // (y, x)
        const float y = (yx.x + 1.0f) * 0.5f * (float)(H - 1);
        const float x = (yx.y + 1.0f) * 0.5f * (float)(W - 1);
        const float y0f = floorf(y);
        const float x0f = floorf(x);
        swy[tid] = y - y0f;
        swx[tid] = x - x0f;
        const int y0 = iclamp((int)y0f,     0, H - 1);
        const int y1 = iclamp((int)y0f + 1, 0, H - 1);
        const int x0 = iclamp((int)x0f,     0, W - 1);
        const int x1 = iclamp((int)x0f + 1, 0, W - 1);
        int4 d;
        d.x = y0 * W + x0;   // v00
        d.y = y0 * W + x1;   // v01
        d.z = y1 * W + x0;   // v10
        d.w = y1 * W + x1;   // v11
        sidx[tid] = d;
    }

    // Drain this wave's async copies, then barrier so all waves see the tile.
    wait_async0();
    __syncthreads();

    // 784 outputs per block; gathers hit LDS, stores are coalesced + NT
    // (output is write-once, never re-read).
    float* obase = out + (size_t)(r * C + c0) * P;
    for (int j = tid; j < CPB * P; j += BLK) {
        const int cl = j / P;
        const int p  = j - cl * P;
        const int4  d  = sidx[p];
        const float wy = swy[p];
        const float wx = swx[p];
        const float* m = &smap[cl * MAP];
        const float v00 = m[d.x];
        const float v01 = m[d.y];
        const float v10 = m[d.z];
        const float v11 = m[d.w];
        const float top = fmaf(wx, v01 - v00, v00);
        const float bot = fmaf(wx, v11 - v10, v10);
        __builtin_nontemporal_store(fmaf(wy, bot - top, top), &obase[j]);
    }
}

extern "C" void kernel_launch(void* const* d_in, const int* in_sizes, int n_in,
                              void* d_out, int out_size, void* d_ws, size_t ws_size,
                              hipStream_t stream) {
    const float* feat = (const float*)d_in[0];   // [128,512,37,37] f32
    const float* grid = (const float*)d_in[1];   // [128,14,14,2]   f32
    float* out = (float*)d_out;                  // [128,512,14,14] f32

    const int nblocks = R * (C / CPB);           // 16384
    roicrop_bilinear_kernel<<<nblocks, BLK, 0, stream>>>(feat, grid, out);
}